// GNNModel_17686675324959
// MI455X (gfx1250) — compile-verified
//
#include <hip/hip_runtime.h>
#include <hip/hip_bf16.h>

typedef __attribute__((ext_vector_type(2))) float v2f;
typedef __attribute__((ext_vector_type(8))) float v8f;

#define FIN  64
#define HID  128
#define FOUT 64

// ---------------------------------------------------------------------------
// Degree accumulation: one thread per (edge ∪ self-loop) endpoint.
// deg[dst] += 1 for every edge, plus self loops.
// ---------------------------------------------------------------------------
__global__ void degree_kernel(const long long* __restrict__ ei, float* __restrict__ deg,
                              int E, int total) {
    int i = blockIdx.x * blockDim.x + threadIdx.x;
    if (i >= total) return;
    int d = (i < E) ? (int)ei[(size_t)E + i] : (i - E);
    atomicAdd(&deg[d], 1.0f);
}

// dinv = deg > 0 ? rsqrt(deg) : 0
__global__ void rsqrt_kernel(const float* __restrict__ deg, float* __restrict__ dinv, int n) {
    int i = blockIdx.x * blockDim.x + threadIdx.x;
    if (i >= n) return;
    float v = deg[i];
    dinv[i] = (v > 0.0f) ? __frsqrt_rn(v) : 0.0f;
}

// ---------------------------------------------------------------------------
// f32 WMMA GEMM: C[M,Ncols] = A[M,K] * B[K,Ncols]
// One wave per 16x16 output tile; K stepped by 4 via V_WMMA_F32_16X16X4_F32.
// M, K, Ncols all multiples of 16 (M=100000=6250*16) -> no tails, EXEC all 1s.
// A frag (16x4 f32, 2 VGPR): lanes 0-15 row M=lane hold K=k0,k0+1;
//                            lanes 16-31 row M=lane-16 hold K=k0+2,k0+3.
// B frag (4x16 f32, 2 VGPR): lanes 0-15 col N=lane hold K=k0,k0+1;
//                            lanes 16-31 col N=lane-16 hold K=k0+2,k0+3.
// C/D (16x16 f32, 8 VGPR): VGPR r -> (M = r + (lane>=16 ? 8:0), N = lane&15).
// ---------------------------------------------------------------------------
__global__ void gemm_wmma_f32(const float* __restrict__ A, const float* __restrict__ B,
                              float* __restrict__ C, int M, int K, int Ncols) {
    const int lane    = threadIdx.x & 31;
    const int wid     = blockIdx.x * (blockDim.x >> 5) + (threadIdx.x >> 5);
    const int tilesN  = Ncols >> 4;
    const int nTiles  = (M >> 4) * tilesN;
    if (wid >= nTiles) return;           // wave-uniform guard: EXEC stays all-ones

    const int mt = wid / tilesN;
    const int nt = wid - mt * tilesN;

    const int halfSel = lane >> 4;       // 0: lanes 0-15, 1: lanes 16-31
    const int kOff    = halfSel * 2;     // K sub-offset 0 or 2
    const int row     = (mt << 4) + (lane & 15);
    const int col     = (nt << 4) + (lane & 15);

    v8f acc = {};
    const float* arow = A + (size_t)row * K;
    for (int k0 = 0; k0 < K; k0 += 4) {
        // A: contiguous K pair -> single 8B load (aligned: k0%4==0, kOff in {0,2})
        const float2 av = *(const float2*)(arow + k0 + kOff);
        v2f a; a.x = av.x; a.y = av.y;
        // B: two rows of W at this column
        v2f b;
        b.x = B[(size_t)(k0 + kOff)     * Ncols + col];
        b.y = B[(size_t)(k0 + kOff + 1) * Ncols + col];
        acc = __builtin_amdgcn_wmma_f32_16x16x4_f32(
            /*neg_a=*/false, a, /*neg_b=*/false, b,
            /*c_mod=*/(short)0, acc, /*reuse_a=*/false, /*reuse_b=*/false);
    }

    const int rbase = (mt << 4) + (halfSel ? 8 : 0);
    float* cptr = C + (size_t)rbase * Ncols + col;
#pragma unroll
    for (int r = 0; r < 8; ++r) cptr[(size_t)r * Ncols] = acc[r];
}

// ---------------------------------------------------------------------------
// Edge scatter, F=128: one wave per edge, lane owns 4 contiguous features.
// agg[dst] += h[src] * (dinv[src]*dinv[dst]);  self-loops appended after E.
// Fully coalesced 512B gather per wave; global_atomic_add_f32 scatter (L2).
// ---------------------------------------------------------------------------
__global__ void scatter128_kernel(const float* __restrict__ h, const long long* __restrict__ ei,
                                  const float* __restrict__ dinv, float* __restrict__ agg,
                                  int E, int total) {
    int wid  = blockIdx.x * (blockDim.x >> 5) + (threadIdx.x >> 5);
    if (wid >= total) return;
    int lane = threadIdx.x & 31;

    int s, d;
    if (wid < E) { s = (int)ei[wid]; d = (int)ei[(size_t)E + wid]; }
    else         { s = d = wid - E; }
    float nrm = dinv[s] * dinv[d];

    const float4 v = *(const float4*)(h + (size_t)s * HID + lane * 4);
    float* p = agg + (size_t)d * HID + lane * 4;
    atomicAdd(p + 0, v.x * nrm);
    atomicAdd(p + 1, v.y * nrm);
    atomicAdd(p + 2, v.z * nrm);
    atomicAdd(p + 3, v.w * nrm);
}

// Edge scatter, F=64: one wave per edge, lane owns 2 contiguous features.
__global__ void scatter64_kernel(const float* __restrict__ h, const long long* __restrict__ ei,
                                 const float* __restrict__ dinv, float* __restrict__ out,
                                 int E, int total) {
    int wid  = blockIdx.x * (blockDim.x >> 5) + (threadIdx.x >> 5);
    if (wid >= total) return;
    int lane = threadIdx.x & 31;

    int s, d;
    if (wid < E) { s = (int)ei[wid]; d = (int)ei[(size_t)E + wid]; }
    else         { s = d = wid - E; }
    float nrm = dinv[s] * dinv[d];

    const float2 v = *(const float2*)(h + (size_t)s * FOUT + lane * 2);
    float* p = out + (size_t)d * FOUT + lane * 2;
    atomicAdd(p + 0, v.x * nrm);
    atomicAdd(p + 1, v.y * nrm);
}

// z = relu(agg + b1[f])  (in place), F = 128
__global__ void bias_relu_kernel(float* __restrict__ z, const float* __restrict__ b, size_t n) {
    size_t i = (size_t)blockIdx.x * blockDim.x + threadIdx.x;
    if (i >= n) return;
    float v = z[i] + b[i & (HID - 1)];
    z[i] = v > 0.0f ? v : 0.0f;
}

// out[i] = b2[i % 64]  (scatter2 then accumulates on top)
__global__ void init_out_kernel(float* __restrict__ out, const float* __restrict__ b, size_t n) {
    size_t i = (size_t)blockIdx.x * blockDim.x + threadIdx.x;
    if (i >= n) return;
    out[i] = b[i & (FOUT - 1)];
}

extern "C" void kernel_launch(void* const* d_in, const int* in_sizes, int n_in,
                              void* d_out, int out_size, void* d_ws, size_t ws_size,
                              hipStream_t stream) {
    const float*     x  = (const float*)d_in[0];       // [N, 64]
    const long long* ei = (const long long*)d_in[1];   // [2, E] int64
    const float*     W1 = (const float*)d_in[2];       // [64, 128]
    const float*     b1 = (const float*)d_in[3];       // [128]
    const float*     W2 = (const float*)d_in[4];       // [128, 64]
    const float*     b2 = (const float*)d_in[5];       // [64]
    float*           out = (float*)d_out;              // [N, 64]

    const int N = in_sizes[0] / FIN;
    const int E = in_sizes[1] / 2;
    const int totalEdges = E + N;                      // edges + self loops

    // workspace layout
    float* deg  = (float*)d_ws;                        // N
    float* dinv = deg + N;                             // N
    float* h    = dinv + N;                            // N*128 (gemm1 out; reused for gemm2 out)
    float* agg  = h + (size_t)N * HID;                 // N*128 (layer-1 aggregate / relu)

    // zero the accumulation buffers every call (graph-replay safe)
    hipMemsetAsync(deg, 0, (size_t)N * sizeof(float), stream);
    hipMemsetAsync(agg, 0, (size_t)N * HID * sizeof(float), stream);

    // degrees -> dinv
    degree_kernel<<<(totalEdges + 255) / 256, 256, 0, stream>>>(ei, deg, E, totalEdges);
    rsqrt_kernel<<<(N + 255) / 256, 256, 0, stream>>>(deg, dinv, N);

    // layer 1: h = x @ W1 (WMMA f32), scatter-agg, bias+relu
    {
        int nTiles = (N >> 4) * (HID >> 4);            // 6250 * 8
        gemm_wmma_f32<<<(nTiles + 7) / 8, 256, 0, stream>>>(x, W1, h, N, FIN, HID);
        int wavesBlocks = (totalEdges * 32 + 255) / 256;
        scatter128_kernel<<<wavesBlocks, 256, 0, stream>>>(h, ei, dinv, agg, E, totalEdges);
        size_t n1 = (size_t)N * HID;
        bias_relu_kernel<<<(unsigned)((n1 + 255) / 256), 256, 0, stream>>>(agg, b1, n1);
    }

    // layer 2: h2 = z @ W2 (WMMA f32, reuse h buffer), out = b2 + scatter
    {
        int nTiles = (N >> 4) * (FOUT >> 4);           // 6250 * 4
        gemm_wmma_f32<<<(nTiles + 7) / 8, 256, 0, stream>>>(agg, W2, h, N, HID, FOUT);
        size_t n2 = (size_t)N * FOUT;
        init_out_kernel<<<(unsigned)((n2 + 255) / 256), 256, 0, stream>>>(out, b2, n2);
        int wavesBlocks = (totalEdges * 32 + 255) / 256;
        scatter64_kernel<<<wavesBlocks, 256, 0, stream>>>(h, ei, dinv, out, E, totalEdges);
    }
}